// KNNTorch_76699525972413
// MI455X (gfx1250) — compile-verified
//
#include <hip/hip_runtime.h>
#include <math.h>

typedef float v2f __attribute__((ext_vector_type(2)));
typedef float v8f __attribute__((ext_vector_type(8)));
typedef unsigned int u32x4 __attribute__((ext_vector_type(4)));
typedef int i32x4 __attribute__((ext_vector_type(4)));
typedef int i32x8 __attribute__((ext_vector_type(8)));

#define DIM   64
#define KKEEP 10
#define BIGF  3.4e38f
#define WAVES 8                 // 8 waves x 16 queries = 128 queries / block
#define LROW  68                // LDS row stride (floats): 64 + 4 pad -> bank-conflict-free
#define LBUF  (16 * LROW)       // one staged 16-row ref chunk

#if defined(__gfx1250__) && __has_builtin(__builtin_amdgcn_tensor_load_to_lds)
#define HAVE_TDM 1
#else
#define HAVE_TDM 0
#endif

// Insert v into ascending sorted array arr[0..KKEEP-1] if it beats the max.
__device__ __forceinline__ void topk_insert(float* arr, float v) {
    if (v < arr[KKEEP - 1]) {
        arr[KKEEP - 1] = v;
#pragma unroll
        for (int i = KKEEP - 1; i >= 1; --i) {
            float lo = fminf(arr[i], arr[i - 1]);
            float hi = fmaxf(arr[i], arr[i - 1]);
            arr[i - 1] = lo;
            arr[i]     = hi;
        }
    }
}

#if HAVE_TDM
// Issue one TDM load: 16 rows x 64 f32 of x_ref starting at row rbase -> LDS,
// padded +4 DWORDs per 64-DWORD row (LDS row stride = 68 floats).
// tensor_dim1 = rows remaining from tile start so OOB rows read as zero.
__device__ __forceinline__ void tdm_load_chunk(const float* R, long long rbase,
                                               int M, unsigned lds_off) {
    long long remain = (long long)M - rbase;
    if (remain <= 0) return;                       // fully OOB chunk: skip
    unsigned rem32 = (unsigned)remain;
    unsigned long long ga =
        (unsigned long long)(const void*)R + (unsigned long long)rbase * (DIM * 4);

    u32x4 g0;
    g0[0] = 1u;                                    // count=1, is_restore=0, no gather
    g0[1] = lds_off;                               // lds_addr (bytes)
    g0[2] = (unsigned)(ga & 0xFFFFFFFFull);        // global_addr[31:0]
    g0[3] = (unsigned)((ga >> 32) & 0x01FFFFFFull) // global_addr[56:32]
          | 0x80000000u;                           // type=2 ("image")

    i32x8 g1;
    g1[0] = (2 << 16)                              // data_size = 4 bytes
          | (1 << 20)                              // pad_enable
          | (5 << 22)                              // pad_interval: 64 DWORDs
          | (3 << 25);                             // pad_amount: 4 DWORDs
    g1[1] = (int)(((unsigned)DIM & 0xFFFFu) << 16);        // tensor_dim0[15:0] @63:48
    g1[2] = (int)((rem32 & 0xFFFFu) << 16);                // dim0 hi=0 | tensor_dim1 lo
    g1[3] = (int)(((rem32 >> 16) & 0xFFFFu)                // tensor_dim1 hi
                  | ((unsigned)DIM << 16));                // tile_dim0 = 64
    g1[4] = 16;                                            // tile_dim1 = 16 rows
    g1[5] = DIM;                                           // tensor_dim0_stride = 64
    g1[6] = 0;
    g1[7] = 0;

    i32x4 z4 = {0, 0, 0, 0};
#if __clang_major__ >= 23
    i32x8 z8 = {0, 0, 0, 0, 0, 0, 0, 0};
    __builtin_amdgcn_tensor_load_to_lds(g0, g1, z4, z4, z8, 0);
#else
    __builtin_amdgcn_tensor_load_to_lds(g0, g1, z4, z4, 0);
#endif
}
#endif

// Phase 1: 8 waves share TDM-staged ref chunks in LDS; each wave runs a
// 16x16 f32 WMMA tile chain + in-register top-10 on s = ||r||^2 - 2<x,r>.
__global__ void __launch_bounds__(32 * WAVES)
knn_phase1(const float* __restrict__ X, const float* __restrict__ R,
           float* __restrict__ cand_out, int N, int M, int S, int chunks) {
    __shared__ float sbuf[2][LBUF];                    // double-buffered ref chunk
    __shared__ float cand[WAVES][16][16][KKEEP];      // per-wave merge scratch

    const int lane = threadIdx.x & 31;
    const int wave = threadIdx.x >> 5;
    const int kh   = lane >> 4;                       // K-half (fragment layout)
    const int nn   = lane & 15;
    const int qbase = blockIdx.x * (16 * WAVES) + wave * 16;
    const int slice = blockIdx.y;
    const long long rbase0 = (long long)slice * chunks * 16;

    // A fragments: this wave's X tile [16 x 64] (32-bit A 16x4 layout).
    v2f afrag[16];
    {
        int row = qbase + nn;
        if (row >= N) row = N - 1;
        const float* xr = X + (long long)row * DIM;
#pragma unroll
        for (int i = 0; i < 16; ++i)
            afrag[i] = *reinterpret_cast<const v2f*>(xr + 4 * i + 2 * kh);
    }

    float lists[8][KKEEP];
#pragma unroll
    for (int j = 0; j < 8; ++j)
#pragma unroll
        for (int t = 0; t < KKEEP; ++t) lists[j][t] = BIGF;

#if HAVE_TDM
    const unsigned lds0 = (unsigned)(unsigned long long)&sbuf[0][0];
    const unsigned lds1 = (unsigned)(unsigned long long)&sbuf[1][0];
    if (wave == 0) {
        tdm_load_chunk(R, rbase0, M, lds0);           // prologue: stage chunk 0
        __builtin_amdgcn_s_wait_tensorcnt(0);
    }
    __syncthreads();
#endif

    for (int c = 0; c < chunks; ++c) {
        const long long rbase = rbase0 + (long long)c * 16;
#if HAVE_TDM
        if (wave == 0 && (c + 1) < chunks)            // async: stage next chunk
            tdm_load_chunk(R, rbase + 16, M, ((c + 1) & 1) ? lds1 : lds0);
        const float* sb = sbuf[c & 1];
#else
        __syncthreads();
        {   // cooperative staging: 256 threads x 4 floats = 16x64 tile
            int t = (int)threadIdx.x;
            int row = t >> 4, col = (t & 15) * 4;
            long long rr = rbase + row;
            if (rr >= M) rr = M - 1;
            const float4 v = *reinterpret_cast<const float4*>(R + rr * DIM + col);
            float* d = &sbuf[0][row * LROW + col];
            d[0] = v.x; d[1] = v.y; d[2] = v.z; d[3] = v.w;
        }
        __syncthreads();
        const float* sb = sbuf[0];
#endif

        // B fragments from LDS (conflict-free thanks to LROW=68).
        v2f bfrag[16];
#pragma unroll
        for (int i = 0; i < 16; ++i)
            bfrag[i] = *reinterpret_cast<const v2f*>(sb + nn * LROW + 4 * i + 2 * kh);

        // ||r||^2: per-lane half, combined across K-halves.
        float r2p = 0.0f;
#pragma unroll
        for (int i = 0; i < 16; ++i)
            r2p = fmaf(bfrag[i].x, bfrag[i].x,
                       fmaf(bfrag[i].y, bfrag[i].y, r2p));
        float r2 = r2p + __shfl_xor(r2p, 16, 32);

        // 16 chained f32 WMMAs: full K=64 contraction for the 16x16 tile.
        v8f acc = {};
#pragma unroll
        for (int i = 0; i < 16; ++i)
            acc = __builtin_amdgcn_wmma_f32_16x16x4_f32(
                false, afrag[i], false, bfrag[i], (short)0, acc, false, false);

        // Epilogue: top-10 insert per query row (OOB refs filtered here).
        const bool valid = (rbase + nn) < (long long)M;
#pragma unroll
        for (int j = 0; j < 8; ++j) {
            float s = fmaf(-2.0f, acc[j], r2);
            if (valid) topk_insert(lists[j], s);
        }

#if HAVE_TDM
        if (wave == 0) __builtin_amdgcn_s_wait_tensorcnt(0);
        __syncthreads();
#endif
    }

    // Cross-lane merge per query via per-wave LDS region.
#pragma unroll
    for (int j = 0; j < 8; ++j) {
        int ql = j + 8 * kh;
#pragma unroll
        for (int t = 0; t < KKEEP; ++t) cand[wave][ql][nn][t] = lists[j][t];
    }
    __syncthreads();

    if (lane < 16) {
        float arr[KKEEP];
#pragma unroll
        for (int t = 0; t < KKEEP; ++t) arr[t] = BIGF;
        for (int n = 0; n < 16; ++n)
            for (int t = 0; t < KKEEP; ++t)
                topk_insert(arr, cand[wave][lane][n][t]);
        int q = qbase + lane;
        if (q < N) {
            float* dst = cand_out + ((long long)q * S + slice) * KKEEP;
#pragma unroll
            for (int t = 0; t < KKEEP; ++t) dst[t] = arr[t];
        }
    }
}

// Phase 2: per-query merge of S*KKEEP candidates; add ||x||^2, clamp, sqrt.
__global__ void __launch_bounds__(256)
knn_phase2(const float* __restrict__ X, const float* __restrict__ cand_in,
           const int* __restrict__ kptr, float* __restrict__ out,
           int N, int S) {
    int q = blockIdx.x * blockDim.x + threadIdx.x;
    if (q >= N) return;

    float arr[KKEEP];
#pragma unroll
    for (int t = 0; t < KKEEP; ++t) arr[t] = BIGF;

    const float* src = cand_in + (long long)q * S * KKEEP;
    for (int i = 0; i < S * KKEEP; ++i) topk_insert(arr, src[i]);

    float x2 = 0.0f;
    const float* xr = X + (long long)q * DIM;
#pragma unroll
    for (int d = 0; d < DIM; ++d) x2 = fmaf(xr[d], xr[d], x2);

    int k = kptr ? kptr[0] : KKEEP;
    if (k < 1) k = 1;
    if (k > KKEEP) k = KKEEP;

    out[q] = sqrtf(fmaxf(arr[k - 1] + x2, 0.0f));
}

extern "C" void kernel_launch(void* const* d_in, const int* in_sizes, int n_in,
                              void* d_out, int out_size, void* d_ws, size_t ws_size,
                              hipStream_t stream) {
    const float* X = (const float*)d_in[0];
    const float* R = (const float*)d_in[1];
    const int* kptr = (n_in > 2) ? (const int*)d_in[2] : nullptr;

    const int N = in_sizes[0] / DIM;   // 2048 queries
    const int M = in_sizes[1] / DIM;   // 100000 refs

    // Ref slices: (qtiles x S) blocks; shrink S if workspace is tight.
    int S = 64;
    while (S > 1 && (size_t)N * S * KKEEP * sizeof(float) > ws_size) S >>= 1;

    const int chunks = (M + S * 16 - 1) / (S * 16);
    const int qtiles = (N + (16 * WAVES) - 1) / (16 * WAVES);

    dim3 g1(qtiles, S);
    knn_phase1<<<g1, 32 * WAVES, 0, stream>>>(X, R, (float*)d_ws, N, M, S,
                                              chunks);

    const int threads = 256;
    const int blocks = (N + threads - 1) / threads;
    knn_phase2<<<blocks, threads, 0, stream>>>(X, (const float*)d_ws, kptr,
                                               (float*)d_out, N, S);
}